// CausalAggregator_18614388261380
// MI455X (gfx1250) — compile-verified
//
#include <hip/hip_runtime.h>

typedef __attribute__((ext_vector_type(2))) float v2f;
typedef __attribute__((ext_vector_type(8))) float v8f;
typedef __attribute__((ext_vector_type(4))) int   v4i;

constexpr int kN = 4096;
constexpr int kD = 256;
constexpr int kO = 256;

// ---------------------------------------------------------------------------
// CDNA5 async global->LDS copy path (ASYNCcnt-tracked), guarded: falls back
// to regular load + ds_store staging if the builtins are absent.
// Probe-derived signature: (addrspace(1) v4i* src, addrspace(3) v4i* dst,
//                           imm offset, imm cpol)
// ---------------------------------------------------------------------------
#if defined(__HIP_DEVICE_COMPILE__) && \
    __has_builtin(__builtin_amdgcn_global_load_async_to_lds_b128)
#define HAVE_ASYNC_LDS 1
#endif

__device__ inline void copy16_to_lds(const float* g, float* l) {
#if defined(HAVE_ASYNC_LDS)
  __builtin_amdgcn_global_load_async_to_lds_b128(
      (__attribute__((address_space(1))) v4i*)g,
      (__attribute__((address_space(3))) v4i*)l, /*offset=*/0, /*cpol=*/0);
#else
  *(float4*)l = *(const float4*)g;
#endif
}

template <int Ncnt>
__device__ inline void async_wait() {
#if defined(HAVE_ASYNC_LDS)
#if __has_builtin(__builtin_amdgcn_s_wait_asynccnt)
  __builtin_amdgcn_s_wait_asynccnt(Ncnt);
#else
  asm volatile("s_wait_asynccnt %0" ::"i"(Ncnt) : "memory");
#endif
#endif
}

// ---------------------------------------------------------------------------
// Zero the adj_agg scratch (4096 x 3 floats). Must run every launch.
// ---------------------------------------------------------------------------
__global__ void zero_kernel(float* __restrict__ p, int n) {
  int i = blockIdx.x * blockDim.x + threadIdx.x;
  if (i < n) p[i] = 0.0f;
}

// ---------------------------------------------------------------------------
// Kernel 1: Y = X @ Wf   (4096x256 @ 256x256), fp32 WMMA 16x16x4.
// One wave per 16x16 output tile; tiny kernel (0.54 GFLOP), global-fed.
// ---------------------------------------------------------------------------
__global__ __launch_bounds__(256) void xwf_kernel(const float* __restrict__ X,
                                                  const float* __restrict__ Wf,
                                                  float* __restrict__ Y) {
  const int lane  = threadIdx.x & 31;
  const int wave  = threadIdx.x >> 5;
  const int tile  = blockIdx.x * 8 + wave;   // 4096 tiles total
  const int m0    = (tile >> 4) * 16;
  const int n0    = (tile & 15) * 16;
  const int row   = lane & 15;
  const int khalf = (lane >> 4) * 2;

  v8f acc = {};
  for (int k = 0; k < kD; k += 4) {
    const int k0 = k + khalf;
    v2f a, b;
    a.x = X[(m0 + row) * kD + k0];
    a.y = X[(m0 + row) * kD + k0 + 1];
    b.x = Wf[k0 * kO + n0 + row];
    b.y = Wf[(k0 + 1) * kO + n0 + row];
    acc = __builtin_amdgcn_wmma_f32_16x16x4_f32(false, a, false, b,
                                                (short)0, acc, false, false);
  }
  const int r0  = (lane < 16) ? 0 : 8;
  const int col = lane & 15;
#pragma unroll
  for (int v = 0; v < 8; ++v)
    Y[(m0 + r0 + v) * kO + n0 + col] = acc[v];
}

// ---------------------------------------------------------------------------
// Kernel 2: adj_agg[y,c] = sum_n Beta[n,y] * adjacency[n,y,c]
// Streaming, HBM-bound (201MB + 64MB => ~11.4us at 23.3 TB/s). Coalesced
// along y; 16 y-blocks x 16 n-chunks; fp32 atomics into small buffer.
// ---------------------------------------------------------------------------
__global__ __launch_bounds__(256) void adjagg_kernel(const float* __restrict__ Beta,
                                                     const float* __restrict__ Adj,
                                                     float* __restrict__ AdjAgg) {
  const int y    = (blockIdx.x & 15) * 256 + threadIdx.x;
  const int nbeg = (blockIdx.x >> 4) * 256;
  float a0 = 0.f, a1 = 0.f, a2 = 0.f;
  for (int n = nbeg; n < nbeg + 256; ++n) {
    const float b = Beta[(size_t)n * kN + y];
    const float* ap = Adj + ((size_t)n * kN + y) * 3;
    a0 += b * ap[0];
    a1 += b * ap[1];
    a2 += b * ap[2];
  }
  atomicAdd(&AdjAgg[y * 3 + 0], a0);
  atomicAdd(&AdjAgg[y * 3 + 1], a1);
  atomicAdd(&AdjAgg[y * 3 + 2], a2);
}

// ---------------------------------------------------------------------------
// Kernel 3: out = Beta^T @ Y + adj_agg @ Wa + bias   (the 8.6 GFLOP GEMM)
//
// Block = 64y x 64o output tile, 8 waves (4 y-waves x 2 o-waves), each wave
// a 16x32 strip with 2 f32 accumulators sharing the A fragment.
// K is chunked by 32; A chunk (32k x 64y of Beta^T) and B chunk (32k x 64o
// of Y) are staged in LDS, double-buffered (40KB of the 320KB WGP LDS),
// via async global->LDS b128 copies. LDS rows padded to 80 floats so the
// two lane-halves of a WMMA fragment hit disjoint bank groups
// (2*80 mod 64 = 32).
// ---------------------------------------------------------------------------
constexpr int kKC   = 32;           // k-chunk
constexpr int kLRow = 80;           // padded LDS row (floats)
constexpr int kBufF = kKC * kLRow;  // 2560 floats per A or B buffer

__device__ inline void stage_chunk(const float* __restrict__ Beta,
                                   const float* __restrict__ Y,
                                   int kbase, int y0, int o0,
                                   float* __restrict__ lA,
                                   float* __restrict__ lB, int t) {
  const int r = t >> 4;            // 0..15
  const int s = (t & 15) * 4;      // 0,4,..,60
  // A tile: Beta rows k (n index), cols y0..y0+63  (contiguous 64 floats/row)
  copy16_to_lds(Beta + (size_t)(kbase + r) * kN + y0 + s, lA + r * kLRow + s);
  copy16_to_lds(Beta + (size_t)(kbase + r + 16) * kN + y0 + s,
                lA + (r + 16) * kLRow + s);
  // B tile: Y rows k, cols o0..o0+63
  copy16_to_lds(Y + (kbase + r) * kO + o0 + s, lB + r * kLRow + s);
  copy16_to_lds(Y + (kbase + r + 16) * kO + o0 + s, lB + (r + 16) * kLRow + s);
}

__device__ inline void compute_chunk(const float* __restrict__ lA,
                                     const float* __restrict__ lB,
                                     int aoff, int boff, int row, int khalf,
                                     v8f& acc0, v8f& acc1) {
#pragma unroll
  for (int kk = 0; kk < kKC; kk += 4) {
    const int k0 = kk + khalf;
    v2f a, b0, b1;
    a.x  = lA[k0 * kLRow + aoff + row];
    a.y  = lA[(k0 + 1) * kLRow + aoff + row];
    b0.x = lB[k0 * kLRow + boff + row];
    b0.y = lB[(k0 + 1) * kLRow + boff + row];
    b1.x = lB[k0 * kLRow + boff + 16 + row];
    b1.y = lB[(k0 + 1) * kLRow + boff + 16 + row];
    acc0 = __builtin_amdgcn_wmma_f32_16x16x4_f32(false, a, false, b0,
                                                 (short)0, acc0, false, false);
    acc1 = __builtin_amdgcn_wmma_f32_16x16x4_f32(false, a, false, b1,
                                                 (short)0, acc1, false, false);
  }
}

__global__ __launch_bounds__(256) void btyo_kernel(const float* __restrict__ Beta,
                                                   const float* __restrict__ Y,
                                                   const float* __restrict__ AdjAgg,
                                                   const float* __restrict__ Wa,
                                                   const float* __restrict__ bias,
                                                   float* __restrict__ out) {
  __shared__ float smem[4 * kBufF];  // {A0,B0,A1,B1} = 40KB

  const int t     = threadIdx.x;
  const int lane  = t & 31;
  const int wave  = t >> 5;
  const int y0    = (blockIdx.x >> 2) * 64;  // 64 y-tiles
  const int o0    = (blockIdx.x & 3) * 64;   // 4 o-tiles
  const int aoff  = (wave & 3) * 16;         // y offset of this wave
  const int boff  = (wave >> 2) * 32;        // o offset of this wave
  const int row   = lane & 15;
  const int khalf = (lane >> 4) * 2;

  v8f acc0 = {}, acc1 = {};

  stage_chunk(Beta, Y, 0, y0, o0, smem, smem + kBufF, t);

  constexpr int NCH = kN / kKC;  // 128 chunks
  for (int c = 0; c < NCH; ++c) {
    const int cur = c & 1;
    float* Acur = smem + cur * (2 * kBufF);
    float* Bcur = Acur + kBufF;
    if (c + 1 < NCH) {
      float* Anxt = smem + (cur ^ 1) * (2 * kBufF);
      stage_chunk(Beta, Y, (c + 1) * kKC, y0, o0, Anxt, Anxt + kBufF, t);
      async_wait<4>();  // the 4 older (chunk c) async ops are complete
    } else {
      async_wait<0>();
    }
    __syncthreads();
    compute_chunk(Acur, Bcur, aoff, boff, row, khalf, acc0, acc1);
    __syncthreads();  // all waves done reading before buffer is re-staged
  }

  // Epilogue: += adj_agg @ Wa + bias, write 16x32 strip.
  const int r0  = (lane < 16) ? 0 : 8;
  const int col = lane & 15;
  const int yw  = y0 + aoff;
  const int n0  = o0 + boff;
#pragma unroll
  for (int v = 0; v < 8; ++v) {
    const int m  = yw + r0 + v;
    const float g0 = AdjAgg[m * 3 + 0];
    const float g1 = AdjAgg[m * 3 + 1];
    const float g2 = AdjAgg[m * 3 + 2];
    const int c0 = n0 + col;
    const int c1 = n0 + 16 + col;
    out[m * kO + c0] = acc0[v] + g0 * Wa[c0] + g1 * Wa[kO + c0]
                     + g2 * Wa[2 * kO + c0] + bias[c0];
    out[m * kO + c1] = acc1[v] + g0 * Wa[c1] + g1 * Wa[kO + c1]
                     + g2 * Wa[2 * kO + c1] + bias[c1];
  }
}

// ---------------------------------------------------------------------------
// Launcher. d_in order: adjacency, input_feature, Beta, weight, bias.
// ws layout: Y (4096*256 f32) | adj_agg (4096*3 f32)  ~= 4.25 MB.
// ---------------------------------------------------------------------------
extern "C" void kernel_launch(void* const* d_in, const int* in_sizes, int n_in,
                              void* d_out, int out_size, void* d_ws, size_t ws_size,
                              hipStream_t stream) {
  const float* adjacency = (const float*)d_in[0];
  const float* X         = (const float*)d_in[1];
  const float* Beta      = (const float*)d_in[2];
  const float* weight    = (const float*)d_in[3];
  const float* bias      = (const float*)d_in[4];
  const float* Wf        = weight;              // (256, 256)
  const float* Wa        = weight + kD * kO;    // (3, 256)

  float* Y      = (float*)d_ws;
  float* AdjAgg = Y + (size_t)kN * kO;

  zero_kernel<<<(kN * 3 + 255) / 256, 256, 0, stream>>>(AdjAgg, kN * 3);
  xwf_kernel<<<512, 256, 0, stream>>>(X, Wf, Y);
  adjagg_kernel<<<256, 256, 0, stream>>>(Beta, adjacency, AdjAgg);
  btyo_kernel<<<256, 256, 0, stream>>>(Beta, Y, AdjAgg, Wa, bias, (float*)d_out);
}